// MultiHeadGATLayer_22153441313023
// MI455X (gfx1250) — compile-verified
//
#include <hip/hip_runtime.h>
#include <math.h>
#include <stdint.h>

#define N_NODES 50000
#define N_EDGES 800000
#define IN_F    128
#define HEADS   4
#define OUT_F   32
#define OUT_TOT (HEADS * OUT_F)   // 128
#define LRELU   0.2f

typedef float v2f __attribute__((ext_vector_type(2)));
typedef float v8f __attribute__((ext_vector_type(8)));
typedef unsigned int u32x4 __attribute__((ext_vector_type(4)));
typedef int i32x4 __attribute__((ext_vector_type(4)));
typedef int i32x8 __attribute__((ext_vector_type(8)));

#if defined(__has_builtin)
#  if __has_builtin(__builtin_amdgcn_tensor_load_to_lds)
#    define HAS_TDM 1
#  else
#    define HAS_TDM 0
#  endif
#else
#  define HAS_TDM 0
#endif

#if HAS_TDM
#  if __has_builtin(__builtin_amdgcn_s_wait_tensorcnt)
#    define WAIT_TENSORCNT0() __builtin_amdgcn_s_wait_tensorcnt(0)
#  else
#    define WAIT_TENSORCNT0() asm volatile("s_wait_tensorcnt 0x0" ::: "memory")
#  endif

// 1-D contiguous global->LDS DMA via the Tensor Data Mover.
// D# built per CDNA5 ISA:
//   group0: count=1 | lds_addr | global_addr[31:0] | global_addr[56:32],type=2
//   group1: data_size=4B, tensor_dim0 = tile_dim0 = ndwords (one row, no pad)
//   groups 2/3 (+extra words): zero (<=2D tensor)
// This toolchain's builtin takes 6 args (g0, g1, g2, g3, extra, cpol).
__device__ __forceinline__ void tdm_copy_to_lds(const void* gsrc, unsigned lds_off,
                                                unsigned ndwords)
{
    const uint64_t ga = (uint64_t)(uintptr_t)gsrc;
    u32x4 g0;
    g0[0] = 1u;                                               // count=1 (valid)
    g0[1] = lds_off;                                          // lds_addr (bytes)
    g0[2] = (unsigned)ga;                                     // global_addr[31:0]
    g0[3] = (unsigned)((ga >> 32) & 0x01FFFFFFu) | 0x80000000u; // [56:32] | type=2

    i32x8 g1;
    g1[0] = (2 << 16);                         // data_size = 2 -> 4 bytes
    g1[1] = (int)((ndwords & 0xFFFFu) << 16);  // tensor_dim0 lo16 @ bits 63:48
    g1[2] = (int)(ndwords >> 16);              // tensor_dim0 hi16 @ bits 79:64
    g1[3] = (int)((ndwords & 0xFFFFu) << 16);  // tile_dim0 @ bits 127:112
    g1[4] = 0;                                 // tile_dim1=0, tile_dim2=0 (1-D)
    g1[5] = (int)ndwords;                      // dim0_stride (unused for 1-D)
    g1[6] = 0;
    g1[7] = 0;

    const i32x4 gz4 = {0, 0, 0, 0};
    const i32x8 gz8 = {0, 0, 0, 0, 0, 0, 0, 0};
    __builtin_amdgcn_tensor_load_to_lds(g0, g1, gz4, gz4, gz8, 0);
}
#endif

// ---------------------------------------------------------------------------
// Kernel 1: h[N,128] = x[N,128] @ Wcat[128,128] via V_WMMA_F32_16X16X4_F32.
// W (64KB) staged once per block into LDS by the TDM; each wave owns one
// 16-row M-tile and produces all 8 column tiles (8 WMMAs per A-fragment,
// x streamed from HBM exactly once).
// ---------------------------------------------------------------------------
__global__ __launch_bounds__(256) void gat_gemm_wmma(
    const float* __restrict__ x,     // [N, IN_F]
    const float* __restrict__ W,     // [HEADS, IN_F, OUT_F] (= 64KB)
    float* __restrict__ h)           // [N, OUT_TOT]
{
    __shared__ float sW[IN_F * OUT_TOT];     // 16384 floats = 64KB, copy of W

#if HAS_TDM
    if ((threadIdx.x >> 5) == 0) {           // wave 0 issues the DMA once
        tdm_copy_to_lds(W, (unsigned)(uintptr_t)sW, IN_F * OUT_TOT);
        WAIT_TENSORCNT0();
    }
#else
    for (int i = threadIdx.x; i < IN_F * OUT_TOT; i += 256) sW[i] = W[i];
#endif
    __syncthreads();                         // publish LDS to all 8 waves

    const int lane   = threadIdx.x & 31;
    const int laneM  = lane & 15;            // M row (A) / N col (B,C)
    const int laneHi = lane >> 4;            // 0 or 1
    const int mTile  = blockIdx.x * 8 + (threadIdx.x >> 5);
    if (mTile * 16 >= N_NODES) return;       // uniform per wave; barrier already done

    const float* __restrict__ xrow = x + (size_t)(mTile * 16 + laneM) * IN_F;

    // B fragment base (dword index in sW) per column tile for this lane:
    // col = nt*16 + laneM ; sW index = head*4096 + k*32 + o
    int bcol[8];
    #pragma unroll
    for (int nt = 0; nt < 8; ++nt) {
        const int c = nt * 16 + laneM;
        bcol[nt] = (c >> 5) * (IN_F * OUT_F) + (c & 31);
    }

    v8f acc[8] = {};
    for (int k0 = 0; k0 < IN_F; k0 += 4) {
        const int kk = k0 + laneHi * 2;
        // A 16x4 f32: VGPR j holds K = laneHi*2 + j  (contiguous -> b64 load)
        v2f a;
        a.x = xrow[kk];
        a.y = xrow[kk + 1];
        #pragma unroll
        for (int nt = 0; nt < 8; ++nt) {
            v2f b;
            b.x = sW[bcol[nt] + kk * OUT_F];
            b.y = sW[bcol[nt] + (kk + 1) * OUT_F];
            acc[nt] = __builtin_amdgcn_wmma_f32_16x16x4_f32(
                false, a, false, b, (short)0, acc[nt], false, false);
        }
    }

    // C/D 16x16 f32: VGPR r -> M = r + 8*laneHi, N = nt*16 + laneM
    #pragma unroll
    for (int nt = 0; nt < 8; ++nt) {
        #pragma unroll
        for (int r = 0; r < 8; ++r) {
            const int m = mTile * 16 + r + laneHi * 8;
            h[(size_t)m * OUT_TOT + nt * 16 + laneM] = acc[nt][r];
        }
    }
}

// ---------------------------------------------------------------------------
// Kernel 2: per-(node, head) attention logits al_src/al_dst = <h, a>
// ---------------------------------------------------------------------------
__global__ __launch_bounds__(256) void gat_al(
    const float* __restrict__ h,
    const float* __restrict__ a_src, const float* __restrict__ a_dst,
    float* __restrict__ al_s, float* __restrict__ al_d)
{
    const int idx = blockIdx.x * blockDim.x + threadIdx.x;   // n*HEADS + head
    if (idx >= N_NODES * HEADS) return;
    const int head = idx & (HEADS - 1);
    const float* __restrict__ hp = h + (size_t)(idx >> 2) * OUT_TOT + head * OUT_F;
    const float* __restrict__ as = a_src + head * OUT_F;
    const float* __restrict__ ad = a_dst + head * OUT_F;
    float s = 0.f, d = 0.f;
    #pragma unroll
    for (int f = 0; f < OUT_F; ++f) {
        const float v = hp[f];
        s = fmaf(v, as[f], s);
        d = fmaf(v, ad[f], d);
    }
    al_s[idx] = s;
    al_d[idx] = d;
}

// ---------------------------------------------------------------------------
// Kernel 3: init m = -inf, denom = 0, out = 0
// ---------------------------------------------------------------------------
__global__ __launch_bounds__(256) void gat_init(
    float* __restrict__ m, float* __restrict__ denom, float* __restrict__ out)
{
    const int idx = blockIdx.x * blockDim.x + threadIdx.x;
    if (idx < N_NODES * HEADS) {
        m[idx] = -INFINITY;
        denom[idx] = 0.f;
    }
    if (idx < N_NODES * OUT_TOT) out[idx] = 0.f;
}

// Sign-aware float atomic max built on integer atomics (init value is -inf;
// monotone in float order for both sign paths, correct for mixed streams).
__device__ __forceinline__ void atomicMaxF(float* addr, float v) {
    if (v >= 0.f) atomicMax((int*)addr, __float_as_int(v));
    else          atomicMin((unsigned int*)addr, __float_as_uint(v));
}

__device__ __forceinline__ float lrelu(float v) {
    return v > 0.f ? v : LRELU * v;
}

// ---------------------------------------------------------------------------
// Kernel 4: segment max of leaky-relu logits over destination nodes
// ---------------------------------------------------------------------------
__global__ __launch_bounds__(256) void gat_segmax(
    const int* __restrict__ src, const int* __restrict__ dst,
    const float* __restrict__ al_s, const float* __restrict__ al_d,
    float* __restrict__ m)
{
    const int idx = blockIdx.x * blockDim.x + threadIdx.x;   // e*HEADS + head
    if (idx >= N_EDGES * HEADS) return;
    const int e = idx >> 2, head = idx & (HEADS - 1);
    const int s = src[e], d = dst[e];
    const float ev = lrelu(al_s[s * HEADS + head] + al_d[d * HEADS + head]);
    atomicMaxF(&m[d * HEADS + head], ev);
}

// ---------------------------------------------------------------------------
// Kernel 5: ex = exp(e - m[dst]); denom[dst] += ex  (ex cached for pass 3)
// ---------------------------------------------------------------------------
__global__ __launch_bounds__(256) void gat_segsum(
    const int* __restrict__ src, const int* __restrict__ dst,
    const float* __restrict__ al_s, const float* __restrict__ al_d,
    const float* __restrict__ m,
    float* __restrict__ ex, float* __restrict__ denom)
{
    const int idx = blockIdx.x * blockDim.x + threadIdx.x;   // e*HEADS + head
    if (idx >= N_EDGES * HEADS) return;
    const int e = idx >> 2, head = idx & (HEADS - 1);
    const int s = src[e], d = dst[e];
    const float ev = lrelu(al_s[s * HEADS + head] + al_d[d * HEADS + head]);
    const float xv = __expf(ev - m[d * HEADS + head]);
    ex[idx] = xv;
    atomicAdd(&denom[d * HEADS + head], xv);
}

// ---------------------------------------------------------------------------
// Kernel 6: out[dst, c] += (ex / denom) * h[src, c]   (one thread per edge*128)
// h and out both fit in the 192MB L2, so the random 512B row traffic stays
// on-chip; consecutive threads cover one contiguous h row -> coalesced.
// ---------------------------------------------------------------------------
__global__ __launch_bounds__(256) void gat_scatter(
    const int* __restrict__ src, const int* __restrict__ dst,
    const float* __restrict__ h, const float* __restrict__ ex,
    const float* __restrict__ denom, float* __restrict__ out)
{
    const long long t = (long long)blockIdx.x * blockDim.x + threadIdx.x;
    if (t >= (long long)N_EDGES * OUT_TOT) return;
    const int e = (int)(t >> 7);
    const int c = (int)(t & (OUT_TOT - 1));
    const int head = c >> 5;
    const int s = src[e], d = dst[e];
    const float w = ex[e * HEADS + head] / (denom[d * HEADS + head] + 1e-16f);
    atomicAdd(&out[d * OUT_TOT + c], w * h[(size_t)s * OUT_TOT + c]);
}

// ---------------------------------------------------------------------------
// Kernel 7: in-place ELU epilogue
// ---------------------------------------------------------------------------
__global__ __launch_bounds__(256) void gat_elu(float* __restrict__ out)
{
    const int idx = blockIdx.x * blockDim.x + threadIdx.x;
    if (idx >= N_NODES * OUT_TOT) return;
    const float v = out[idx];
    out[idx] = v > 0.f ? v : (__expf(v) - 1.f);
}

// ---------------------------------------------------------------------------
extern "C" void kernel_launch(void* const* d_in, const int* in_sizes, int n_in,
                              void* d_out, int out_size, void* d_ws, size_t ws_size,
                              hipStream_t stream) {
    const float* x          = (const float*)d_in[0];          // [N, 128]
    const int*   edge_index = (const int*)d_in[1];            // [2, E]
    const float* W          = (const float*)d_in[2];          // [H, 128, 32]
    const float* a_src      = (const float*)d_in[3];          // [H, 32]
    const float* a_dst      = (const float*)d_in[4];          // [H, 32]
    float*       out        = (float*)d_out;                  // [N, 128]

    const int* src = edge_index;              // row 0
    const int* dst = edge_index + N_EDGES;    // row 1

    // Workspace carve-up (floats): h | al_s | al_d | m | denom | ex  (~41.6MB)
    float* ws    = (float*)d_ws;
    float* h     = ws;                                   // N*128 = 6.40M
    float* al_s  = h    + (size_t)N_NODES * OUT_TOT;     // N*H   = 0.20M
    float* al_d  = al_s + (size_t)N_NODES * HEADS;
    float* m     = al_d + (size_t)N_NODES * HEADS;
    float* denom = m    + (size_t)N_NODES * HEADS;
    float* ex    = denom+ (size_t)N_NODES * HEADS;       // E*H   = 3.20M

    const int B = 256;

    // 1) WMMA GEMM: 3125 M-tiles, 8 waves/block -> 391 blocks
    gat_gemm_wmma<<<(N_NODES / 16 + 7) / 8, B, 0, stream>>>(x, W, h);

    // 2) attention logits
    gat_al<<<(N_NODES * HEADS + B - 1) / B, B, 0, stream>>>(h, a_src, a_dst, al_s, al_d);

    // 3) init m/denom/out
    gat_init<<<(N_NODES * OUT_TOT + B - 1) / B, B, 0, stream>>>(m, denom, out);

    // 4) segment max
    gat_segmax<<<(N_EDGES * HEADS + B - 1) / B, B, 0, stream>>>(src, dst, al_s, al_d, m);

    // 5) exp + segment sum
    gat_segsum<<<(N_EDGES * HEADS + B - 1) / B, B, 0, stream>>>(src, dst, al_s, al_d, m, ex, denom);

    // 6) weighted scatter-add aggregation (102.4M threads, L2-resident traffic)
    const long long tot = (long long)N_EDGES * OUT_TOT;
    gat_scatter<<<(unsigned)((tot + B - 1) / B), B, 0, stream>>>(src, dst, h, ex, denom, out);

    // 7) ELU
    gat_elu<<<(N_NODES * OUT_TOT + B - 1) / B, B, 0, stream>>>(out);
}